// SemiseparableLayer_53910429499410
// MI455X (gfx1250) — compile-verified
//
#include <hip/hip_runtime.h>

// ---------------------------------------------------------------------------
// Semiseparable layer for MI455X (gfx1250, wave32, WMMA), blocked-scan version.
//
//   causal:     y_k  = x C_k^T + u_k D_k^T ;  x <- x A_k^T + u_k B_k^T
//   anticausal: y_k += z G_k^T             ;  z <- z E_k^T + u_k F_k^T (k dec)
//
// Stages split into NC chunks of LCH: K1 computes per-chunk local scans from
// zero plus chunk transition matrices M_c / N_c; K2 propagates chunk-boundary
// carries; K3 re-runs local scans from true initial states and emits y once
// (both directions accumulated in LDS). All GEMMs: V_WMMA_F32_16X16X4_F32.
// ---------------------------------------------------------------------------

typedef float v2f __attribute__((ext_vector_type(2)));
typedef float v4f __attribute__((ext_vector_type(4)));
typedef float v8f __attribute__((ext_vector_type(8)));

#define BATCH   4096
#define INPUT   8192
#define IO      16
#define DSTATE  32
#define NST     512            // INPUT / IO
#define NTILES  (BATCH / 16)   // 256
#define NC      16             // chunks over the stage axis
#define LCH     (NST / NC)     // 32 stages per chunk
#define SLOT    512            // 16x32 state tile, fragment-order dump

static __device__ __forceinline__ v8f wmma4(v2f a, v2f b, v8f c) {
  // D(16x16,f32) = A(16x4,f32) * B(4x16,f32) + C
  return __builtin_amdgcn_wmma_f32_16x16x4_f32(
      false, a, false, b, (short)0, c, false, false);
}

// ============================ K1: local scans ==============================
__global__ __launch_bounds__(32)
void k1_local(const float* __restrict__ U,  const float* __restrict__ A,
              const float* __restrict__ Bm, const float* __restrict__ Em,
              const float* __restrict__ Fm,
              float* __restrict__ XE, float* __restrict__ ZE,
              float* __restrict__ MC, float* __restrict__ NCm)
{
  __shared__ __align__(16) float S[2048];
  const int lane = threadIdx.x & 31;
  const int half = lane >> 4;
  const int n    = lane & 15;
  const int bid  = blockIdx.x;

  if (bid < NTILES * NC) {
    // ---- (chunk, tile) local scans starting from zero state ----
    float* Xlds = S;         // 16x32 row-major state
    float* Ulds = S + 512;   // 16x16 row-major u tile
    const int c = bid / NTILES, t = bid % NTILES;
    const int rowBase = t * 16, k0 = c * LCH;
    const int urow = lane >> 1, ucol = (lane & 1) * 8;

    // causal
    for (int i = lane; i < 512; i += 32) Xlds[i] = 0.0f;
    v8f x0 = {}, x1 = {};
    for (int j = 0; j < LCH; ++j) {
      const int k = k0 + j;
      const float* usrc = U + (size_t)(rowBase + urow) * INPUT + k * IO + ucol;
      *(v4f*)&Ulds[urow * IO + ucol]     = *(const v4f*)(usrc);
      *(v4f*)&Ulds[urow * IO + ucol + 4] = *(const v4f*)(usrc + 4);
      if (j + 1 < LCH) __builtin_prefetch(usrc + IO, 0, 0);
      const float* Ak = A  + (size_t)k * DSTATE * DSTATE;
      const float* Bk = Bm + (size_t)k * DSTATE * IO;
      v8f xn0 = {}, xn1 = {};
#pragma unroll
      for (int cc = 0; cc < 8; ++cc) {
        const int ko = 4 * cc + 2 * half;
        v2f a  = *(const v2f*)&Xlds[n * DSTATE + ko];
        v2f b0 = *(const v2f*)(Ak + n * DSTATE + ko);
        v2f b1 = *(const v2f*)(Ak + (16 + n) * DSTATE + ko);
        xn0 = wmma4(a, b0, xn0); xn1 = wmma4(a, b1, xn1);
      }
#pragma unroll
      for (int cc = 0; cc < 4; ++cc) {
        const int ko = 4 * cc + 2 * half;
        v2f a  = *(const v2f*)&Ulds[n * IO + ko];
        v2f b0 = *(const v2f*)(Bk + n * IO + ko);
        v2f b1 = *(const v2f*)(Bk + (16 + n) * IO + ko);
        xn0 = wmma4(a, b0, xn0); xn1 = wmma4(a, b1, xn1);
      }
#pragma unroll
      for (int i = 0; i < 8; ++i) {
        const int m = i + 8 * half;
        Xlds[m * DSTATE + n]      = xn0[i];
        Xlds[m * DSTATE + 16 + n] = xn1[i];
      }
      x0 = xn0; x1 = xn1;
    }
    { float* slot = XE + ((size_t)c * NTILES + t) * SLOT + lane * 16;
      *(v8f*)slot = x0; *(v8f*)(slot + 8) = x1; }

    // anticausal
    for (int i = lane; i < 512; i += 32) Xlds[i] = 0.0f;
    v8f z0 = {}, z1 = {};
    for (int j = LCH - 1; j >= 0; --j) {
      const int k = k0 + j;
      const float* usrc = U + (size_t)(rowBase + urow) * INPUT + k * IO + ucol;
      *(v4f*)&Ulds[urow * IO + ucol]     = *(const v4f*)(usrc);
      *(v4f*)&Ulds[urow * IO + ucol + 4] = *(const v4f*)(usrc + 4);
      const float* Ek = Em + (size_t)k * DSTATE * DSTATE;
      const float* Fk = Fm + (size_t)k * DSTATE * IO;
      v8f zn0 = {}, zn1 = {};
#pragma unroll
      for (int cc = 0; cc < 8; ++cc) {
        const int ko = 4 * cc + 2 * half;
        v2f a  = *(const v2f*)&Xlds[n * DSTATE + ko];
        v2f b0 = *(const v2f*)(Ek + n * DSTATE + ko);
        v2f b1 = *(const v2f*)(Ek + (16 + n) * DSTATE + ko);
        zn0 = wmma4(a, b0, zn0); zn1 = wmma4(a, b1, zn1);
      }
#pragma unroll
      for (int cc = 0; cc < 4; ++cc) {
        const int ko = 4 * cc + 2 * half;
        v2f a  = *(const v2f*)&Ulds[n * IO + ko];
        v2f b0 = *(const v2f*)(Fk + n * IO + ko);
        v2f b1 = *(const v2f*)(Fk + (16 + n) * IO + ko);
        zn0 = wmma4(a, b0, zn0); zn1 = wmma4(a, b1, zn1);
      }
#pragma unroll
      for (int i = 0; i < 8; ++i) {
        const int m = i + 8 * half;
        Xlds[m * DSTATE + n]      = zn0[i];
        Xlds[m * DSTATE + 16 + n] = zn1[i];
      }
      z0 = zn0; z1 = zn1;
    }
    { float* slot = ZE + ((size_t)c * NTILES + t) * SLOT + lane * 16;
      *(v8f*)slot = z0; *(v8f*)(slot + 8) = z1; }

  } else {
    // ---- chunk transition matrices: M_c = A^T chain, N_c = E^T chain ----
    const int c = bid - NTILES * NC;
    float* Mlds = S;            // 32x32 row-major
    // M_c : M <- M * A_k^T, k ascending
    for (int i = lane; i < 1024; i += 32)
      Mlds[i] = ((i >> 5) == (i & 31)) ? 1.0f : 0.0f;
    for (int j = 0; j < LCH; ++j) {
      const float* Ak = A + (size_t)(c * LCH + j) * DSTATE * DSTATE;
      v8f m00 = {}, m01 = {}, m10 = {}, m11 = {};
#pragma unroll
      for (int cc = 0; cc < 8; ++cc) {
        const int ko = 4 * cc + 2 * half;
        v2f a0 = *(const v2f*)&Mlds[n * DSTATE + ko];
        v2f a1 = *(const v2f*)&Mlds[(16 + n) * DSTATE + ko];
        v2f b0 = *(const v2f*)(Ak + n * DSTATE + ko);
        v2f b1 = *(const v2f*)(Ak + (16 + n) * DSTATE + ko);
        m00 = wmma4(a0, b0, m00); m01 = wmma4(a0, b1, m01);
        m10 = wmma4(a1, b0, m10); m11 = wmma4(a1, b1, m11);
      }
#pragma unroll
      for (int i = 0; i < 8; ++i) {
        const int r = i + 8 * half;
        Mlds[r * DSTATE + n]             = m00[i];
        Mlds[r * DSTATE + 16 + n]        = m01[i];
        Mlds[(16 + r) * DSTATE + n]      = m10[i];
        Mlds[(16 + r) * DSTATE + 16 + n] = m11[i];
      }
    }
    for (int i = lane; i < 1024; i += 32) MC[(size_t)c * 1024 + i] = Mlds[i];

    // N_c : N <- N * E_k^T, k descending
    for (int i = lane; i < 1024; i += 32)
      Mlds[i] = ((i >> 5) == (i & 31)) ? 1.0f : 0.0f;
    for (int j = LCH - 1; j >= 0; --j) {
      const float* Ek = Em + (size_t)(c * LCH + j) * DSTATE * DSTATE;
      v8f m00 = {}, m01 = {}, m10 = {}, m11 = {};
#pragma unroll
      for (int cc = 0; cc < 8; ++cc) {
        const int ko = 4 * cc + 2 * half;
        v2f a0 = *(const v2f*)&Mlds[n * DSTATE + ko];
        v2f a1 = *(const v2f*)&Mlds[(16 + n) * DSTATE + ko];
        v2f b0 = *(const v2f*)(Ek + n * DSTATE + ko);
        v2f b1 = *(const v2f*)(Ek + (16 + n) * DSTATE + ko);
        m00 = wmma4(a0, b0, m00); m01 = wmma4(a0, b1, m01);
        m10 = wmma4(a1, b0, m10); m11 = wmma4(a1, b1, m11);
      }
#pragma unroll
      for (int i = 0; i < 8; ++i) {
        const int r = i + 8 * half;
        Mlds[r * DSTATE + n]             = m00[i];
        Mlds[r * DSTATE + 16 + n]        = m01[i];
        Mlds[(16 + r) * DSTATE + n]      = m10[i];
        Mlds[(16 + r) * DSTATE + 16 + n] = m11[i];
      }
    }
    for (int i = lane; i < 1024; i += 32) NCm[(size_t)c * 1024 + i] = Mlds[i];
  }
}

// ===================== K2: chunk-boundary carry propagation ================
__global__ __launch_bounds__(32)
void k2_carry(const float* __restrict__ MC, const float* __restrict__ NCm,
              float* __restrict__ XE, float* __restrict__ ZE)
{
  __shared__ __align__(16) float Xlds[512];
  const int lane = threadIdx.x & 31;
  const int half = lane >> 4;
  const int n    = lane & 15;
  const int t    = blockIdx.x;

  // causal: x_init(0)=0 ; x_init(c+1) = x_init(c)*M_c + x_end0(c)
  v8f x0 = {}, x1 = {};
  for (int c = 0; c < NC; ++c) {
    float* slot = XE + ((size_t)c * NTILES + t) * SLOT + lane * 16;
    v8f e0 = *(const v8f*)slot;
    v8f e1 = *(const v8f*)(slot + 8);
    *(v8f*)slot = x0; *(v8f*)(slot + 8) = x1;   // replace x_end0 with x_init
    if (c < NC - 1) {
#pragma unroll
      for (int i = 0; i < 8; ++i) {
        const int m = i + 8 * half;
        Xlds[m * DSTATE + n] = x0[i]; Xlds[m * DSTATE + 16 + n] = x1[i];
      }
      const float* Mc = MC + (size_t)c * 1024;
      v8f n0 = e0, n1 = e1;
#pragma unroll
      for (int cc = 0; cc < 8; ++cc) {
        const int ko = 4 * cc + 2 * half;
        v2f a = *(const v2f*)&Xlds[n * DSTATE + ko];
        v2f b0, b1;
        b0.x = Mc[ko * DSTATE + n];      b0.y = Mc[(ko + 1) * DSTATE + n];
        b1.x = Mc[ko * DSTATE + 16 + n]; b1.y = Mc[(ko + 1) * DSTATE + 16 + n];
        n0 = wmma4(a, b0, n0); n1 = wmma4(a, b1, n1);
      }
      x0 = n0; x1 = n1;
    }
  }

  // anticausal: z_init(NC-1)=0 ; z_init(c-1) = z_init(c)*N_c + z_end0(c)
  v8f z0 = {}, z1 = {};
  for (int c = NC - 1; c >= 0; --c) {
    float* slot = ZE + ((size_t)c * NTILES + t) * SLOT + lane * 16;
    v8f e0 = *(const v8f*)slot;
    v8f e1 = *(const v8f*)(slot + 8);
    *(v8f*)slot = z0; *(v8f*)(slot + 8) = z1;
    if (c > 0) {
#pragma unroll
      for (int i = 0; i < 8; ++i) {
        const int m = i + 8 * half;
        Xlds[m * DSTATE + n] = z0[i]; Xlds[m * DSTATE + 16 + n] = z1[i];
      }
      const float* Nc = NCm + (size_t)c * 1024;
      v8f n0 = e0, n1 = e1;
#pragma unroll
      for (int cc = 0; cc < 8; ++cc) {
        const int ko = 4 * cc + 2 * half;
        v2f a = *(const v2f*)&Xlds[n * DSTATE + ko];
        v2f b0, b1;
        b0.x = Nc[ko * DSTATE + n];      b0.y = Nc[(ko + 1) * DSTATE + n];
        b1.x = Nc[ko * DSTATE + 16 + n]; b1.y = Nc[(ko + 1) * DSTATE + 16 + n];
        n0 = wmma4(a, b0, n0); n1 = wmma4(a, b1, n1);
      }
      z0 = n0; z1 = n1;
    }
  }
}

// ================= K3: final local scans + single-shot y emit ==============
__global__ __launch_bounds__(32)
void k3_final(const float* __restrict__ U,  const float* __restrict__ A,
              const float* __restrict__ Bm, const float* __restrict__ Cm,
              const float* __restrict__ Dm, const float* __restrict__ Em,
              const float* __restrict__ Fm, const float* __restrict__ Gm,
              const float* __restrict__ bias,
              const float* __restrict__ XE, const float* __restrict__ ZE,
              float* __restrict__ out)
{
  __shared__ __align__(16) float Xlds[512];
  __shared__ __align__(16) float Ulds[256];
  __shared__ __align__(16) float Ylds[16 * LCH * IO];   // 32 KB y-tile buffer

  const int lane = threadIdx.x & 31;
  const int half = lane >> 4;
  const int n    = lane & 15;
  const int c = blockIdx.x / NTILES, t = blockIdx.x % NTILES;
  const int rowBase = t * 16, k0 = c * LCH;
  const int urow = lane >> 1, ucol = (lane & 1) * 8;
  const int YW = LCH * IO;  // 512

  // ---- causal from x_init(c) ----
  { const float* slot = XE + ((size_t)c * NTILES + t) * SLOT + lane * 16;
    v8f x0 = *(const v8f*)slot, x1 = *(const v8f*)(slot + 8);
#pragma unroll
    for (int i = 0; i < 8; ++i) {
      const int m = i + 8 * half;
      Xlds[m * DSTATE + n] = x0[i]; Xlds[m * DSTATE + 16 + n] = x1[i];
    } }

  for (int j = 0; j < LCH; ++j) {
    const int k = k0 + j;
    const float* usrc = U + (size_t)(rowBase + urow) * INPUT + k * IO + ucol;
    *(v4f*)&Ulds[urow * IO + ucol]     = *(const v4f*)(usrc);
    *(v4f*)&Ulds[urow * IO + ucol + 4] = *(const v4f*)(usrc + 4);
    if (j + 1 < LCH) __builtin_prefetch(usrc + IO, 0, 0);
    const float* Ak = A  + (size_t)k * DSTATE * DSTATE;
    const float* Bk = Bm + (size_t)k * DSTATE * IO;
    const float* Ck = Cm + (size_t)k * IO * DSTATE;
    const float* Dk = Dm + (size_t)k * IO * IO;
    v8f y = {}, xn0 = {}, xn1 = {};
#pragma unroll
    for (int cc = 0; cc < 8; ++cc) {
      const int ko = 4 * cc + 2 * half;
      v2f a  = *(const v2f*)&Xlds[n * DSTATE + ko];
      v2f bC = *(const v2f*)(Ck + n * DSTATE + ko);
      y = wmma4(a, bC, y);
      v2f b0 = *(const v2f*)(Ak + n * DSTATE + ko);
      v2f b1 = *(const v2f*)(Ak + (16 + n) * DSTATE + ko);
      xn0 = wmma4(a, b0, xn0); xn1 = wmma4(a, b1, xn1);
    }
#pragma unroll
    for (int cc = 0; cc < 4; ++cc) {
      const int ko = 4 * cc + 2 * half;
      v2f a  = *(const v2f*)&Ulds[n * IO + ko];
      v2f bD = *(const v2f*)(Dk + n * IO + ko);
      y = wmma4(a, bD, y);
      v2f b0 = *(const v2f*)(Bk + n * IO + ko);
      v2f b1 = *(const v2f*)(Bk + (16 + n) * IO + ko);
      xn0 = wmma4(a, b0, xn0); xn1 = wmma4(a, b1, xn1);
    }
#pragma unroll
    for (int i = 0; i < 8; ++i) {
      const int m = i + 8 * half;
      Xlds[m * DSTATE + n]      = xn0[i];
      Xlds[m * DSTATE + 16 + n] = xn1[i];
    }
    const float bv = bias[k * IO + n];
#pragma unroll
    for (int i = 0; i < 8; ++i) {
      const int m = i + 8 * half;
      Ylds[m * YW + j * IO + n] = y[i] + bv;
    }
  }

  // ---- anticausal from z_init(c), accumulate into Ylds ----
  { const float* slot = ZE + ((size_t)c * NTILES + t) * SLOT + lane * 16;
    v8f z0 = *(const v8f*)slot, z1 = *(const v8f*)(slot + 8);
#pragma unroll
    for (int i = 0; i < 8; ++i) {
      const int m = i + 8 * half;
      Xlds[m * DSTATE + n] = z0[i]; Xlds[m * DSTATE + 16 + n] = z1[i];
    } }

  for (int j = LCH - 1; j >= 0; --j) {
    const int k = k0 + j;
    const float* usrc = U + (size_t)(rowBase + urow) * INPUT + k * IO + ucol;
    *(v4f*)&Ulds[urow * IO + ucol]     = *(const v4f*)(usrc);
    *(v4f*)&Ulds[urow * IO + ucol + 4] = *(const v4f*)(usrc + 4);
    const float* Ek = Em + (size_t)k * DSTATE * DSTATE;
    const float* Fk = Fm + (size_t)k * DSTATE * IO;
    const float* Gk = Gm + (size_t)k * IO * DSTATE;
    v8f y = {}, zn0 = {}, zn1 = {};
#pragma unroll
    for (int cc = 0; cc < 8; ++cc) {
      const int ko = 4 * cc + 2 * half;
      v2f a  = *(const v2f*)&Xlds[n * DSTATE + ko];
      v2f bG = *(const v2f*)(Gk + n * DSTATE + ko);
      y = wmma4(a, bG, y);
      v2f b0 = *(const v2f*)(Ek + n * DSTATE + ko);
      v2f b1 = *(const v2f*)(Ek + (16 + n) * DSTATE + ko);
      zn0 = wmma4(a, b0, zn0); zn1 = wmma4(a, b1, zn1);
    }
#pragma unroll
    for (int cc = 0; cc < 4; ++cc) {
      const int ko = 4 * cc + 2 * half;
      v2f a  = *(const v2f*)&Ulds[n * IO + ko];
      v2f b0 = *(const v2f*)(Fk + n * IO + ko);
      v2f b1 = *(const v2f*)(Fk + (16 + n) * IO + ko);
      zn0 = wmma4(a, b0, zn0); zn1 = wmma4(a, b1, zn1);
    }
#pragma unroll
    for (int i = 0; i < 8; ++i) {
      const int m = i + 8 * half;
      Xlds[m * DSTATE + n]      = zn0[i];
      Xlds[m * DSTATE + 16 + n] = zn1[i];
    }
#pragma unroll
    for (int i = 0; i < 8; ++i) {
      const int m = i + 8 * half;
      Ylds[m * YW + j * IO + n] += y[i];
    }
  }

  // ---- single coalesced dump of the 16 x 512 y-tile ----
  for (int idx = lane * 4; idx < 16 * YW; idx += 32 * 4) {
    const int r = idx / YW, cc2 = idx % YW;
    *(v4f*)&out[(size_t)(rowBase + r) * INPUT + k0 * IO + cc2] =
        *(const v4f*)&Ylds[idx];
  }
}

// =============== fallback: round-1 monolithic serial-scan kernel ===========
__global__ __launch_bounds__(32)
void semisep_mono(const float* __restrict__ U,  const float* __restrict__ A,
                  const float* __restrict__ Bm, const float* __restrict__ Cm,
                  const float* __restrict__ Dm, const float* __restrict__ Em,
                  const float* __restrict__ Fm, const float* __restrict__ Gm,
                  const float* __restrict__ bias, float* __restrict__ out)
{
  __shared__ __align__(16) float Xlds[512];
  __shared__ __align__(16) float Ulds[256];
  const int lane = threadIdx.x & 31;
  const int half = lane >> 4;
  const int n    = lane & 15;
  const int rowBase = blockIdx.x * 16;
  const int urow = lane >> 1, ucol = (lane & 1) * 8;

  for (int i = lane; i < 512; i += 32) Xlds[i] = 0.0f;
  for (int k = 0; k < NST; ++k) {
    const float* usrc = U + (size_t)(rowBase + urow) * INPUT + k * IO + ucol;
    *(v4f*)&Ulds[urow * IO + ucol]     = *(const v4f*)(usrc);
    *(v4f*)&Ulds[urow * IO + ucol + 4] = *(const v4f*)(usrc + 4);
    const float* Ak = A  + (size_t)k * DSTATE * DSTATE;
    const float* Bk = Bm + (size_t)k * DSTATE * IO;
    const float* Ck = Cm + (size_t)k * IO * DSTATE;
    const float* Dk = Dm + (size_t)k * IO * IO;
    v8f y = {}, xn0 = {}, xn1 = {};
#pragma unroll
    for (int cc = 0; cc < 8; ++cc) {
      const int ko = 4 * cc + 2 * half;
      v2f a  = *(const v2f*)&Xlds[n * DSTATE + ko];
      v2f bC = *(const v2f*)(Ck + n * DSTATE + ko);
      y = wmma4(a, bC, y);
      v2f b0 = *(const v2f*)(Ak + n * DSTATE + ko);
      v2f b1 = *(const v2f*)(Ak + (16 + n) * DSTATE + ko);
      xn0 = wmma4(a, b0, xn0); xn1 = wmma4(a, b1, xn1);
    }
#pragma unroll
    for (int cc = 0; cc < 4; ++cc) {
      const int ko = 4 * cc + 2 * half;
      v2f a  = *(const v2f*)&Ulds[n * IO + ko];
      v2f bD = *(const v2f*)(Dk + n * IO + ko);
      y = wmma4(a, bD, y);
      v2f b0 = *(const v2f*)(Bk + n * IO + ko);
      v2f b1 = *(const v2f*)(Bk + (16 + n) * IO + ko);
      xn0 = wmma4(a, b0, xn0); xn1 = wmma4(a, b1, xn1);
    }
#pragma unroll
    for (int i = 0; i < 8; ++i) {
      const int m = i + 8 * half;
      Xlds[m * DSTATE + n] = xn0[i]; Xlds[m * DSTATE + 16 + n] = xn1[i];
    }
    const float bv = bias[k * IO + n];
#pragma unroll
    for (int i = 0; i < 8; ++i) {
      const int m = i + 8 * half;
      out[(size_t)(rowBase + m) * INPUT + k * IO + n] = y[i] + bv;
    }
  }

  for (int i = lane; i < 512; i += 32) Xlds[i] = 0.0f;
  for (int k = NST - 1; k >= 0; --k) {
    const float* usrc = U + (size_t)(rowBase + urow) * INPUT + k * IO + ucol;
    *(v4f*)&Ulds[urow * IO + ucol]     = *(const v4f*)(usrc);
    *(v4f*)&Ulds[urow * IO + ucol + 4] = *(const v4f*)(usrc + 4);
    const float* Ek = Em + (size_t)k * DSTATE * DSTATE;
    const float* Fk = Fm + (size_t)k * DSTATE * IO;
    const float* Gk = Gm + (size_t)k * IO * DSTATE;
    v8f y = {}, zn0 = {}, zn1 = {};
#pragma unroll
    for (int cc = 0; cc < 8; ++cc) {
      const int ko = 4 * cc + 2 * half;
      v2f a  = *(const v2f*)&Xlds[n * DSTATE + ko];
      v2f bG = *(const v2f*)(Gk + n * DSTATE + ko);
      y = wmma4(a, bG, y);
      v2f b0 = *(const v2f*)(Ek + n * DSTATE + ko);
      v2f b1 = *(const v2f*)(Ek + (16 + n) * DSTATE + ko);
      zn0 = wmma4(a, b0, zn0); zn1 = wmma4(a, b1, zn1);
    }
#pragma unroll
    for (int cc = 0; cc < 4; ++cc) {
      const int ko = 4 * cc + 2 * half;
      v2f a  = *(const v2f*)&Ulds[n * IO + ko];
      v2f b0 = *(const v2f*)(Fk + n * IO + ko);
      v2f b1 = *(const v2f*)(Fk + (16 + n) * IO + ko);
      zn0 = wmma4(a, b0, zn0); zn1 = wmma4(a, b1, zn1);
    }
#pragma unroll
    for (int i = 0; i < 8; ++i) {
      const int m = i + 8 * half;
      Xlds[m * DSTATE + n] = zn0[i]; Xlds[m * DSTATE + 16 + n] = zn1[i];
    }
#pragma unroll
    for (int i = 0; i < 8; ++i) {
      const int m = i + 8 * half;
      float* p = out + (size_t)(rowBase + m) * INPUT + k * IO + n;
      *p = *p + y[i];
    }
  }
}

// ===========================================================================
extern "C" void kernel_launch(void* const* d_in, const int* in_sizes, int n_in,
                              void* d_out, int out_size, void* d_ws, size_t ws_size,
                              hipStream_t stream) {
  const float* U    = (const float*)d_in[0];
  const float* A    = (const float*)d_in[1];
  const float* B    = (const float*)d_in[2];
  const float* C    = (const float*)d_in[3];
  const float* D    = (const float*)d_in[4];
  const float* E    = (const float*)d_in[5];
  const float* F    = (const float*)d_in[6];
  const float* G    = (const float*)d_in[7];
  const float* bias = (const float*)d_in[8];
  float* out = (float*)d_out;

  const size_t SEG  = (size_t)NC * NTILES * SLOT;       // x/z state dumps
  const size_t MATS = (size_t)NC * 1024;                // M_c / N_c
  const size_t need = (2 * SEG + 2 * MATS) * sizeof(float);

  if (ws_size >= need) {
    float* XE  = (float*)d_ws;
    float* ZE  = XE + SEG;
    float* MC  = ZE + SEG;
    float* NCm = MC + MATS;
    hipLaunchKernelGGL(k1_local, dim3(NTILES * NC + NC), dim3(32), 0, stream,
                       U, A, B, E, F, XE, ZE, MC, NCm);
    hipLaunchKernelGGL(k2_carry, dim3(NTILES), dim3(32), 0, stream,
                       MC, NCm, XE, ZE);
    hipLaunchKernelGGL(k3_final, dim3(NTILES * NC), dim3(32), 0, stream,
                       U, A, B, C, D, E, F, G, bias, XE, ZE, out);
  } else {
    hipLaunchKernelGGL(semisep_mono, dim3(NTILES), dim3(32), 0, stream,
                       U, A, B, C, D, E, F, G, bias, out);
  }
}